// FasterRCNN_26087631356450
// MI455X (gfx1250) — compile-verified
//
#include <hip/hip_runtime.h>
#include <stdint.h>

typedef unsigned int u32x4 __attribute__((ext_vector_type(4)));
typedef int          i32x4 __attribute__((ext_vector_type(4)));
typedef int          i32x8 __attribute__((ext_vector_type(8)));

#define NGT 64
#define BLOCK 256

// FPN level boundaries (cumulative anchor counts): g = 1024/st, count = 3*g*g
//  L0: st=4  sz=32  g=256 -> 196608
//  L1: st=8  sz=64  g=128 -> 245760
//  L2: st=16 sz=128 g=64  -> 258048
//  L3: st=32 sz=256 g=32  -> 261120
//  L4: st=64 sz=512 g=16  -> 261888

__global__ __launch_bounds__(BLOCK) void rcnn_match_kernel(
    const float* __restrict__ deltas,
    const float* __restrict__ gt,
    float* __restrict__ out,
    int total)
{
    __shared__ __align__(16) float  s_raw[NGT * 4]; // raw gt via TDM (must be first: lds_addr 0)
    __shared__ float4 s_box[NGT];                   // {x1,y1,x2,y2}
    __shared__ float  s_ar[NGT];                    // gt area
    __shared__ float4 s_enc[NGT];                   // {gx,gy,gw,gh}

    const unsigned tid = threadIdx.x;

    // ---- TDM: DMA the 64x4 f32 gt block into LDS (issued by wave 0 only) ----
    if (tid < 32u) {
        unsigned long long ga = (unsigned long long)(uintptr_t)gt;
        // Generic shared pointer: low 32 bits == LDS byte offset (ISA aperture rule)
        unsigned lds_off = (unsigned)(uintptr_t)(&s_raw[0]);

        u32x4 g0;
        g0.x = 1u;                                      // count=1, user-mode descriptor
        g0.y = lds_off;                                 // lds_addr [63:32]
        g0.z = (unsigned)(ga & 0xffffffffull);          // global_addr[31:0]
        g0.w = (unsigned)((ga >> 32) & 0x01ffffffull)   // global_addr[56:32]
             | (2u << 30);                              // type = 2 ("image")

        i32x8 g1;
        g1[0] = (int)(2u << 16);      // workgroup_mask=0, data_size=2 (4 bytes)
        g1[1] = (int)(256u << 16);    // tensor_dim0 = 256 elements (lo16 -> [31:16])
        g1[2] = (int)(1u << 16);      // tensor_dim0 hi = 0 | tensor_dim1 = 1
        g1[3] = (int)(256u << 16);    // tensor_dim1 hi = 0 | tile_dim0 = 256
        g1[4] = 1;                    // tile_dim1 = 1, tile_dim2 = 0
        g1[5] = 256;                  // tensor_dim0_stride = 256 (lo32)
        g1[6] = (int)(256u << 16);    // stride0 hi = 0 | tensor_dim1_stride lo16 = 256
        g1[7] = 0;                    // stride1 hi = 0

        i32x4 gz = {0, 0, 0, 0};
#if __clang_major__ >= 23
        i32x8 gz8 = {0, 0, 0, 0, 0, 0, 0, 0};
        __builtin_amdgcn_tensor_load_to_lds(g0, g1, gz, gz, gz8, 0);
#else
        __builtin_amdgcn_tensor_load_to_lds(g0, g1, gz, gz, 0);
#endif
        __builtin_amdgcn_s_wait_tensorcnt(0);
    }
    __syncthreads();

    // ---- derive per-gt quantities once per block (packed for b128 LDS reads) ----
    if (tid < NGT) {
        float4 b = ((const float4*)s_raw)[tid];
        float gw = b.z - b.x, gh = b.w - b.y;
        s_box[tid] = b;
        s_ar[tid]  = gw * gh;
        s_enc[tid] = make_float4(b.x + 0.5f * gw, b.y + 0.5f * gh, gw, gh);
    }
    __syncthreads();

    const unsigned i = blockIdx.x * BLOCK + tid;
    if ((int)i >= total) return;

    // ---- analytic anchor regeneration ----
    unsigned li; unsigned glog; float stf, szf;
    if (i < 196608u)      { li = i;           glog = 8u; stf = 4.0f;  szf = 32.0f;  }
    else if (i < 245760u) { li = i - 196608u; glog = 7u; stf = 8.0f;  szf = 64.0f;  }
    else if (i < 258048u) { li = i - 245760u; glog = 6u; stf = 16.0f; szf = 128.0f; }
    else if (i < 261120u) { li = i - 258048u; glog = 5u; stf = 32.0f; szf = 256.0f; }
    else                  { li = i - 261120u; glog = 4u; stf = 64.0f; szf = 512.0f; }

    unsigned r    = li % 3u;
    unsigned cell = li / 3u;
    unsigned ixc  = cell & ((1u << glog) - 1u);
    unsigned iyc  = cell >> glog;

    float sx = ((float)ixc + 0.5f) * stf;
    float sy = ((float)iyc + 0.5f) * stf;

    const float HM0 = 0.35355339059327373f; // sqrt(0.5)/2
    const float HM2 = 0.70710678118654752f; // sqrt(2)/2
    float hwm = (r == 0u) ? HM0 : ((r == 1u) ? 0.5f : HM2);
    float hhm = (r == 0u) ? HM2 : ((r == 1u) ? 0.5f : HM0);
    float hw = szf * hwm, hh = szf * hhm;   // exact: szf is a power of two

    float ax1 = sx - hw, ay1 = sy - hh, ax2 = sx + hw, ay2 = sy + hh;
    float aw = ax2 - ax1, ah = ay2 - ay1;
    float ax = ax1 + 0.5f * aw, ay = ay1 + 0.5f * ah;

    // ---- decode box ----
    float4 d = ((const float4*)deltas)[i];        // coalesced global_load_b128
    float dw = fminf(d.z, 4.0f);
    float dh = fminf(d.w, 4.0f);
    float px = d.x * aw + ax;
    float py = d.y * ah + ay;
    float pw = expf(dw) * aw;
    float ph = expf(dh) * ah;
    float bx1 = px - 0.5f * pw, by1 = py - 0.5f * ph;
    float bx2 = px + 0.5f * pw, by2 = py + 0.5f * ph;
    float area_a = (bx2 - bx1) * (by2 - by1);

    // ---- IoU max/argmax over 64 gt boxes (LDS b128 broadcast) ----
    float best = -__builtin_inff();
    int bj = 0;
#pragma unroll 8
    for (int j = 0; j < NGT; ++j) {
        float4 b = s_box[j];
        float ix1 = fmaxf(bx1, b.x);
        float iy1 = fmaxf(by1, b.y);
        float ix2 = fminf(bx2, b.z);
        float iy2 = fminf(by2, b.w);
        float iw = fmaxf(ix2 - ix1, 0.0f);
        float ih = fmaxf(iy2 - iy1, 0.0f);
        float inter = iw * ih;
        float iou = inter / (area_a + s_ar[j] - inter + 1e-6f);
        if (iou > best) { best = iou; bj = j; }   // strict '>' keeps first max (jnp.argmax)
    }

    // ---- encode targets vs anchor ----
    float4 e = s_enc[bj];
    float tx = (e.x - ax) / aw;
    float ty = (e.y - ay) / ah;
    float tw = logf(e.z / aw);
    float th = logf(e.w / ah);

    float* o = out + (size_t)i * 9u;
    o[0] = bx1; o[1] = by1; o[2] = bx2; o[3] = by2;
    o[4] = tx;  o[5] = ty;  o[6] = tw;  o[7] = th;
    o[8] = best;
}

extern "C" void kernel_launch(void* const* d_in, const int* in_sizes, int n_in,
                              void* d_out, int out_size, void* d_ws, size_t ws_size,
                              hipStream_t stream) {
    const float* deltas = (const float*)d_in[0];   // (A, 4) f32
    const float* gt     = (const float*)d_in[1];   // (64, 4) f32
    float* out          = (float*)d_out;           // (A, 9) f32

    int total = in_sizes[0] / 4;                   // 261888
    int grid  = (total + BLOCK - 1) / BLOCK;       // 1023
    rcnn_match_kernel<<<grid, BLOCK, 0, stream>>>(deltas, gt, out, total);
}